// DMNN_40879498728939
// MI455X (gfx1250) — compile-verified
//
#include <hip/hip_runtime.h>
#include <cstdint>

typedef float v2f __attribute__((ext_vector_type(2)));
typedef float v8f __attribute__((ext_vector_type(8)));
typedef unsigned int u32x4 __attribute__((ext_vector_type(4)));
typedef int i32x4 __attribute__((ext_vector_type(4)));
typedef int i32x8 __attribute__((ext_vector_type(8)));

#if defined(__HIP_DEVICE_COMPILE__) && __has_builtin(__builtin_amdgcn_tensor_load_to_lds)
#define DMNN_USE_TDM 1
#if __has_include(<hip/amd_detail/amd_gfx1250_TDM.h>)
#define DMNN_TDM_SIX_ARG 1  // therock-10.0 headers -> 6-arg builtin
#endif
#endif

namespace {
constexpr int kBatch = 65536;
constexpr int kC = 2;
constexpr int kK = 512;
constexpr int kD = 64;
constexpr float kEps = 1e-12f;

constexpr int kWaves = 8;
constexpr int kBlock = kWaves * 32;         // 256 threads = 8 wave32
constexpr int kRowsPerBlock = kWaves * 16;  // 128 batch rows per block
constexpr int kTiles = kK / 16;             // 32 dendrite tiles per class
constexpr int kLdsPitch = kD + 4;           // 68-float pitch: kills 16-way bank conflicts

constexpr size_t kSmemBytes = (size_t)kK * kLdsPitch * 4  // centroids (padded)
                            + (size_t)kK * 8              // {c2, radius}
                            + (size_t)kK * 8;             // {W[0,c,k], W[1,c,k]}
}  // namespace

__global__ __launch_bounds__(kBlock) void dmnn_fused(
    const float* __restrict__ inp, const float* __restrict__ cent,
    const float* __restrict__ radii, const float* __restrict__ W,
    const float* __restrict__ bias, float* __restrict__ out) {
  extern __shared__ float smem[];
  float* sC = smem;                                  // [kK][kLdsPitch]
  float2* sMeta = (float2*)(smem + kK * kLdsPitch);  // {c2, radius} per dendrite
  float2* sW = sMeta + kK;                           // {W[0,c,k], W[1,c,k]}

  const int tid = threadIdx.x;
  const int lane = tid & 31;
  const int wave = tid >> 5;
  const int half = lane >> 4;  // which half of the wave
  const int l = lane & 15;
  const int row = blockIdx.x * kRowsPerBlock + wave * 16 + l;

  // ---- B-matrix (4x16 per K-step): lanes 0-15 hold depths {4s,4s+1} of row l,
  //      lanes 16-31 hold depths {4s+2,4s+3}. 16 rows live in 32 VGPRs/wave.
  v2f bmat[16];
  {
    const float* ip = inp + (size_t)row * kD + half * 2;
#pragma unroll
    for (int s = 0; s < 16; ++s) bmat[s] = *(const v2f*)(ip + s * 4);
  }
  float x2p = 0.f;
#pragma unroll
  for (int s = 0; s < 16; ++s)
    x2p = fmaf(bmat[s].x, bmat[s].x, fmaf(bmat[s].y, bmat[s].y, x2p));
  const float x2 = x2p + __shfl_xor(x2p, 16);  // each half holds half the depths

  float logit0 = bias[0] + bias[1];  // sum(b, axis=1)
  float logit1 = bias[2] + bias[3];

  for (int c = 0; c < kC; ++c) {
    __syncthreads();  // previous class fully consumed before restaging

#if defined(DMNN_USE_TDM)
    // ---- TDM: one descriptor moves the whole 512x64 f32 class tile into LDS,
    //      with hardware row padding 64 DWORDs -> +4 DWORDs (68-float pitch).
    if (wave == 0) {
      const uint64_t ga = (uint64_t)(uintptr_t)(cent + (size_t)c * kK * kD);
      const uint32_t ldsa = (uint32_t)(uintptr_t)sC;  // low 32 bits = LDS offset
      const u32x4 g0 = {
          1u,                  // [1:0] count=1 valid descriptor (user mode)
          ldsa,                // [63:32] lds_addr
          (uint32_t)ga,        // [95:64] global_addr[31:0]
          (uint32_t)((ga >> 32) & 0x01FFFFFFu) | (2u << 30)};  // addr[56:32]|type=2
      const i32x8 g1 = {
          (int)((2u << 16)      // data_size = 4 bytes
                | (1u << 20)    // pad_enable
                | (5u << 22)    // pad_interval: every 2^(5+1)=64 DWORDs
                | (3u << 25)),  // pad_amount: 3+1 = 4 DWORDs
          (int)(64u << 16),     // tensor_dim0 = 64  (bits 79:48, low half)
          (int)(512u << 16),    // tensor_dim1 = 512 (bits 111:80, low half)
          (int)(64u << 16),     // tile_dim0 = 64    (bits 127:112)
          (int)512,             // tile_dim1 = 512   (bits 143:128)
          (int)64,              // tensor_dim0_stride = 64 (bits 207:160)
          0, 0};
      const i32x4 gz = {0, 0, 0, 0};
#if defined(DMNN_TDM_SIX_ARG)
      const i32x8 gz8 = {0, 0, 0, 0, 0, 0, 0, 0};
      __builtin_amdgcn_tensor_load_to_lds(g0, g1, gz, gz, gz8, 0);
#else
      __builtin_amdgcn_tensor_load_to_lds(g0, g1, gz, gz, 0);
#endif
      __builtin_amdgcn_s_wait_tensorcnt(0);
    }
    for (int r = tid; r < kK; r += kBlock) {
      sW[r] = make_float2(W[c * kK + r], W[kC * kK + c * kK + r]);
      sMeta[r].y = radii[c * kK + r];
    }
#else
    // ---- fallback staging: padded float4 copies ----
    for (int r = tid; r < kK; r += kBlock) {
      const float4* src = (const float4*)(cent + ((size_t)c * kK + r) * kD);
      float4* dst = (float4*)(sC + r * kLdsPitch);
#pragma unroll
      for (int q = 0; q < kD / 4; ++q) dst[q] = src[q];
      sW[r] = make_float2(W[c * kK + r], W[kC * kK + c * kK + r]);
      sMeta[r].y = radii[c * kK + r];
    }
#endif
    if (c == 0) {  // pull class-1 centroids toward L2 while class 0 computes
      const char* nxt = (const char*)(cent + (size_t)kK * kD);
      for (int i = tid * 128; i < kK * kD * 4; i += kBlock * 128)
        __builtin_prefetch(nxt + i, 0, 1);
    }
    __syncthreads();
    for (int r = tid; r < kK; r += kBlock) {  // ||centroid||^2 from LDS copy
      const float* cp = sC + r * kLdsPitch;
      float s = 0.f;
      for (int d0 = 0; d0 < kD; ++d0) s = fmaf(cp[d0], cp[d0], s);
      sMeta[r].x = s;
    }
    __syncthreads();

    // ---- flash-style online d*softmax(d) over the 512 dendrites of class c
    float mrun = -3.0e38f, Z = 0.f, a0 = 0.f, a1 = 0.f;
    for (int t = 0; t < kTiles; ++t) {
      // Two independent accumulator chains -> 2x WMMA pipeline ILP.
      v8f accA = {0.f, 0.f, 0.f, 0.f, 0.f, 0.f, 0.f, 0.f};
      v8f accB = {0.f, 0.f, 0.f, 0.f, 0.f, 0.f, 0.f, 0.f};
      // A-matrix (16x4): lanes 0-15 row m=t*16+l depths {4s,4s+1};
      //                  lanes 16-31 same rows, depths {4s+2,4s+3}.
      const float* ap = sC + (t * 16 + l) * kLdsPitch + half * 2;
#pragma unroll
      for (int s = 0; s < 16; s += 2) {
        v2f aA = *(const v2f*)(ap + s * 4);
        v2f aB = *(const v2f*)(ap + s * 4 + 4);
        accA = __builtin_amdgcn_wmma_f32_16x16x4_f32(
            false, aA, false, bmat[s], (short)0, accA, false, false);
        accB = __builtin_amdgcn_wmma_f32_16x16x4_f32(
            false, aB, false, bmat[s + 1], (short)0, accB, false, false);
      }
      const v8f acc = accA + accB;
      // D layout: lane holds M = t*16 + half*8 + j for its own batch row N=l.
      const int mb = t * 16 + half * 8;
      float dval[8];
#pragma unroll
      for (int j = 0; j < 8; ++j) {
        float2 meta = sMeta[mb + j];
        float sq = fmaxf(fmaf(-2.f, acc[j], x2 + meta.x), kEps);
        dval[j] = meta.y - __builtin_amdgcn_sqrtf(sq);
      }
      float ml = dval[0];
#pragma unroll
      for (int j = 1; j < 8; ++j) ml = fmaxf(ml, dval[j]);
      const float mnew = fmaxf(mrun, ml);
      const float scale = __expf(mrun - mnew);
      Z *= scale;
      a0 *= scale;
      a1 *= scale;
#pragma unroll
      for (int j = 0; j < 8; ++j) {
        float2 w = sW[mb + j];
        float e = __expf(dval[j] - mnew);
        Z += e;
        float ge = dval[j] * e;  // d * exp(d - m)
        a0 = fmaf(ge, w.x, a0);
        a1 = fmaf(ge, w.y, a1);
      }
      mrun = mnew;
    }
    // ---- merge the two lane-half partial softmax states for this batch row
    const float mo = __shfl_xor(mrun, 16);
    const float Zo = __shfl_xor(Z, 16);
    const float a0o = __shfl_xor(a0, 16);
    const float a1o = __shfl_xor(a1, 16);
    const float mm = fmaxf(mrun, mo);
    const float s1 = __expf(mrun - mm), s2 = __expf(mo - mm);
    const float invZ = 1.0f / fmaf(Z, s1, Zo * s2);
    logit0 = fmaf(fmaf(a0, s1, a0o * s2), invZ, logit0);
    logit1 = fmaf(fmaf(a1, s1, a1o * s2), invZ, logit1);
  }

  // ---- final 2-way softmax; one coalesced float2 store per batch row ----
  const float mx = fmaxf(logit0, logit1);
  const float e0 = __expf(logit0 - mx), e1 = __expf(logit1 - mx);
  const float inv = 1.0f / (e0 + e1);
  if (half == 0) {
    *(float2*)(out + (size_t)row * 2) = make_float2(e0 * inv, e1 * inv);
  }
}

extern "C" void kernel_launch(void* const* d_in, const int* in_sizes, int n_in,
                              void* d_out, int out_size, void* d_ws,
                              size_t ws_size, hipStream_t stream) {
  (void)in_sizes; (void)n_in; (void)out_size; (void)d_ws; (void)ws_size;
  const float* inp = (const float*)d_in[0];
  const float* cent = (const float*)d_in[1];
  const float* radii = (const float*)d_in[2];
  const float* W = (const float*)d_in[3];
  const float* bias = (const float*)d_in[4];
  float* out = (float*)d_out;

  dim3 grid(kBatch / kRowsPerBlock);  // 512 blocks
  dim3 block(kBlock);                 // 256 threads (8 wave32)
  hipLaunchKernelGGL(dmnn_fused, grid, block, kSmemBytes, stream,
                     inp, cent, radii, W, bias, out);
}